// SimpleMamba_74423193305841
// MI455X (gfx1250) — compile-verified
//
#include <hip/hip_runtime.h>
#include <stdint.h>
#include <math.h>

// ---------------------------------------------------------------------------
// Diagonal linear recurrence (simple Mamba) on gfx1250.
//   s_t = A*s_{t-1} + B*x_t ;  y_t = tanh(C*s_t)
// Bandwidth-bound (>=134 MB traffic, ~100 MFLOP) -> chunked parallel scan:
//   pass1: per-(chunk,channel) partial scan (zero init), async-LDS streaming
//   pass2: cross-chunk carry scan (decay A^LC via 5 squarings)
//   pass3: replay chunks from carries, hw v_tanh_f32, non-temporal y stores
// ---------------------------------------------------------------------------

typedef __attribute__((ext_vector_type(4))) float f4;
typedef __attribute__((ext_vector_type(4))) int   v4i;
typedef __attribute__((address_space(1))) v4i* gptr_v4i;   // global (AS1)
typedef __attribute__((address_space(3))) v4i* lptr_v4i;   // LDS (AS3)

#define L_TOT   8192
#define D_DIM   2048
#define G_CHN   256                 // time chunks
#define LC      (L_TOT / G_CHN)     // 32 steps per chunk (2^5)
#define CH4     (D_DIM / 4)         // 512 float4 channel-groups (2^9)
#define TBLOCK  256                 // 8 waves (wave32)
#define TSTEP   4                   // timesteps per async stage
#define NSTG    (LC / TSTEP)        // 8 stages

static_assert(L_TOT % G_CHN == 0, "chunking");
static_assert(LC % TSTEP == 0, "staging");
static_assert((CH4 & (CH4 - 1)) == 0, "pow2 channel groups");

// ---- gfx1250 feature probes (compile-only loop: fall back cleanly) --------
#if defined(__has_builtin)
#  if __has_builtin(__builtin_amdgcn_global_load_async_to_lds_b128) && \
      __has_builtin(__builtin_amdgcn_s_wait_asynccnt)
#    define HAVE_ASYNC_LDS 1
#  endif
#  if __has_builtin(__builtin_amdgcn_tanhf)
#    define TANHF_HW(v) __builtin_amdgcn_tanhf(v)
#  elif __has_builtin(__builtin_amdgcn_tanh_f32)
#    define TANHF_HW(v) __builtin_amdgcn_tanh_f32(v)
#  endif
#  if __has_builtin(__builtin_amdgcn_s_wait_dscnt)
#    define WAIT_DS0() __builtin_amdgcn_s_wait_dscnt(0)
#  endif
#endif
#ifndef WAIT_DS0
#  define WAIT_DS0() asm volatile("s_wait_dscnt 0" ::: "memory")
#endif

__device__ __forceinline__ float hw_tanhf(float v) {
#if defined(TANHF_HW)
  return TANHF_HW(v);
#else
  // gfx1250 VALU transcendental; manual TRANS hazard cover since the
  // compiler cannot see through inline asm.
  float r;
  asm("v_tanh_f32 %0, %1\n\ts_delay_alu instid0(TRANS32_DEP_1)"
      : "=v"(r) : "v"(v));
  return r;
#endif
}

// ---------------------------------------------------------------------------
// Pass 1: per-chunk partial scan with zero initial state.
// thread -> (chunk g, channel-group c4 of 4 channels). Wave reads 512B/step.
// Async double-buffered LDS staging gives 2*TSTEP=8 float4 loads in flight
// per thread (~8 MB chip-wide) without VGPR buffering.
// ---------------------------------------------------------------------------
__global__ __launch_bounds__(TBLOCK) void mamba_partial(
    const float* __restrict__ xg, const float* __restrict__ Ag,
    const float* __restrict__ Bg, float* __restrict__ Sg)
{
  const int tid = threadIdx.x;
  const int gid = blockIdx.x * TBLOCK + tid;
  const int c4  = gid & (CH4 - 1);
  const int g   = gid >> 9;                       // log2(CH4) == 9
  const f4 a = *(const f4*)(Ag + 4 * c4);
  const f4 b = *(const f4*)(Bg + 4 * c4);
  const f4* __restrict__ xrow = (const f4*)xg + (size_t)g * LC * CH4 + c4;
  f4 s = {0.f, 0.f, 0.f, 0.f};

#if defined(HAVE_ASYNC_LDS)
  __shared__ f4 stage[2][TSTEP][TBLOCK];          // 32 KB, per-thread slots
  #pragma unroll
  for (int t = 0; t < TSTEP; ++t)
    __builtin_amdgcn_global_load_async_to_lds_b128(
        (gptr_v4i)(xrow + (size_t)t * CH4),
        (lptr_v4i)&stage[0][t][tid], 0, 0);

  for (int stg = 0; stg < NSTG; ++stg) {
    const int buf = stg & 1;
    if (stg + 1 < NSTG) {
      // Ensure our earlier ds reads of buf^1 completed before overwrite (WAR).
      WAIT_DS0();
      const f4* nxt = xrow + (size_t)(stg + 1) * TSTEP * CH4;
      #pragma unroll
      for (int t = 0; t < TSTEP; ++t)
        __builtin_amdgcn_global_load_async_to_lds_b128(
            (gptr_v4i)(nxt + (size_t)t * CH4),
            (lptr_v4i)&stage[buf ^ 1][t][tid], 0, 0);
      __builtin_amdgcn_s_wait_asynccnt(TSTEP);    // current stage landed (in-order)
    } else {
      __builtin_amdgcn_s_wait_asynccnt(0);
    }
    #pragma unroll
    for (int t = 0; t < TSTEP; ++t) {
      f4 v = stage[buf][t][tid];
      s = a * s + b * v;                          // contracts to v_fma chains
    }
  }
#else
  #pragma unroll 8
  for (int t = 0; t < LC; ++t) {
    f4 v = xrow[(size_t)t * CH4];
    s = a * s + b * v;
  }
#endif
  *(f4*)(Sg + (size_t)g * D_DIM + 4 * c4) = s;
}

// ---------------------------------------------------------------------------
// Pass 2: per-channel scan across the G_CHN chunk sums; rewrites Sg in place
// with the *incoming carry* of each chunk. One thread per channel, coalesced.
// ---------------------------------------------------------------------------
__global__ __launch_bounds__(TBLOCK) void mamba_carry(
    const float* __restrict__ Ag, float* __restrict__ Sg)
{
  const int d = blockIdx.x * TBLOCK + threadIdx.x;
  float p = Ag[d];
  #pragma unroll
  for (int i = 0; i < 5; ++i) p = p * p;          // A^LC, LC = 32 = 2^5
  float c = 0.0f;
  #pragma unroll 4
  for (int g = 0; g < G_CHN; ++g) {
    const float sv = Sg[(size_t)g * D_DIM + d];
    Sg[(size_t)g * D_DIM + d] = c;                // carry INTO chunk g
    c = (g == 0) ? sv : fmaf(p, c, sv);           // avoid inf*0 -> NaN at g==0
  }
}

// ---------------------------------------------------------------------------
// Pass 3: replay each chunk from its carry; y = tanh(C*s) via v_tanh_f32.
// Non-temporal x loads (last use) + non-temporal y stores keep x resident in
// L2 from pass 1 (64 MB < 192 MB L2) and avoid polluting it with y.
// ---------------------------------------------------------------------------
__global__ __launch_bounds__(TBLOCK) void mamba_final(
    const float* __restrict__ xg, const float* __restrict__ Ag,
    const float* __restrict__ Bg, const float* __restrict__ Cg,
    const float* __restrict__ carry, float* __restrict__ yg)
{
  const int gid = blockIdx.x * TBLOCK + threadIdx.x;
  const int c4  = gid & (CH4 - 1);
  const int g   = gid >> 9;
  const f4 a  = *(const f4*)(Ag + 4 * c4);
  const f4 b  = *(const f4*)(Bg + 4 * c4);
  const f4 cc = *(const f4*)(Cg + 4 * c4);
  f4 s = *(const f4*)(carry + (size_t)g * D_DIM + 4 * c4);
  const f4* __restrict__ xrow = (const f4*)xg + (size_t)g * LC * CH4 + c4;
  f4* __restrict__ yrow       = (f4*)yg       + (size_t)g * LC * CH4 + c4;

  #pragma unroll 8
  for (int t = 0; t < LC; ++t) {
    f4 v = __builtin_nontemporal_load(xrow + (size_t)t * CH4);
    s = a * s + b * v;
    f4 o;
    o.x = hw_tanhf(cc.x * s.x);
    o.y = hw_tanhf(cc.y * s.y);
    o.z = hw_tanhf(cc.z * s.z);
    o.w = hw_tanhf(cc.w * s.w);
    __builtin_nontemporal_store(o, yrow + (size_t)t * CH4);
  }
}

// ---------------------------------------------------------------------------
extern "C" void kernel_launch(void* const* d_in, const int* in_sizes, int n_in,
                              void* d_out, int out_size, void* d_ws, size_t ws_size,
                              hipStream_t stream) {
  (void)in_sizes; (void)n_in; (void)out_size; (void)ws_size;
  const float* x = (const float*)d_in[0];
  const float* A = (const float*)d_in[1];
  const float* B = (const float*)d_in[2];
  const float* C = (const float*)d_in[3];
  float* y = (float*)d_out;
  float* S = (float*)d_ws;                        // G_CHN*D_DIM floats = 2 MB

  const int nthreads = CH4 * G_CHN;               // 131072
  dim3 blk(TBLOCK);
  mamba_partial<<<nthreads / TBLOCK, blk, 0, stream>>>(x, A, B, S);
  mamba_carry  <<<D_DIM   / TBLOCK, blk, 0, stream>>>(A, S);
  mamba_final  <<<nthreads / TBLOCK, blk, 0, stream>>>(x, A, B, C, S, y);
}